// TrajectoryEncoder_87737591922948
// MI455X (gfx1250) — compile-verified
//
#include <hip/hip_runtime.h>

// Problem constants (from the reference)
#define BATCH   512
#define SEQ     200
#define SEQP    208      // padded to 13*16 for WMMA M tiling
#define EMB     256
#define CDIM    64
#define NSKILL  300
#define MTILES  13       // SEQP / 16
#define NTILES  16       // EMB / 16

typedef __attribute__((ext_vector_type(2))) float v2f;
typedef __attribute__((ext_vector_type(8))) float v8f;

__global__ __launch_bounds__(256)
void traj_encoder_wmma_kernel(const int*   __restrict__ q,
                              const int*   __restrict__ r,
                              const float* __restrict__ skill_w, const float* __restrict__ skill_b,
                              const float* __restrict__ att_w,   const float* __restrict__ att_b,
                              const float* __restrict__ mast_w,  const float* __restrict__ mast_b,
                              const float* __restrict__ proj_w,  const float* __restrict__ proj_b,
                              float*       __restrict__ out)
{
    __shared__ int    qv[SEQP];
    __shared__ int    rv[SEQP];
    __shared__ float4 xs[SEQP];   // (skill_f, attempts, mastery, valid)
    __shared__ float  wc[CDIM];   // concat(skill_w, att_w, mast_w)
    __shared__ float  bc[CDIM];   // concat(skill_b, att_b, mast_b)
    __shared__ float  pb[EMB];    // proj_b

    const int tid = threadIdx.x;
    const int b   = blockIdx.x;

    // ---------------- Phase 0: stage row + params into LDS ----------------
    if (tid < SEQP) {
        if (tid < SEQ) { qv[tid] = q[b * SEQ + tid]; rv[tid] = r[b * SEQ + tid]; }
        else           { qv[tid] = -1;               rv[tid] = 0; }
    }
    if (tid < CDIM) {
        float w, bb;
        if (tid < 21)      { w = skill_w[tid];      bb = skill_b[tid];      }
        else if (tid < 42) { w = att_w[tid - 21];   bb = att_b[tid - 21];   }
        else               { w = mast_w[tid - 42];  bb = mast_b[tid - 42];  }
        wc[tid] = w; bc[tid] = bb;
    }
    pb[tid] = proj_b[tid];   // blockDim.x == EMB == 256
    __syncthreads();

    // ------- Phase 1: running counts as parallel pairwise prefix ----------
    // attempts[s] = #{ t <= s : valid[t] && q[t]==q[s] }, correct likewise.
    if (tid < SEQP) {
        float4 v4;
        if (tid < SEQ) {
            const int  qs = qv[tid];
            const bool vs = (qs >= 0) && (qs < NSKILL);
            int cnt = 0, cor = 0;
            for (int t = 0; t <= tid; ++t) {
                const int  qt = qv[t];
                const bool m  = (qt == qs) && (qt >= 0) && (qt < NSKILL);
                cnt += m ? 1 : 0;
                cor += m ? rv[t] : 0;
            }
            const float att  = (float)cnt;
            const float mast = (float)cor / fmaxf(att, 1.0f);
            v4 = make_float4((float)qs, att, mast, vs ? 1.0f : 0.0f);
        } else {
            v4 = make_float4(0.f, 0.f, 0.f, 0.f);
        }
        xs[tid] = v4;
    }
    __syncthreads();

    // ---- Phase 2: [SEQP x 64] x [64 x 256] via V_WMMA_F32_16X16X4_F32 ----
    // A layout (16x4 f32): lane holds row m = lane&15; VGPR j holds K = 2*(lane>>4)+j.
    // B layout mirrors A with N = lane&15: VGPR j holds K = 2*(lane>>4)+j.
    // C/D layout: VGPR v holds row m = v + 8*(lane>>4), column n = lane&15.
    const int wave = tid >> 5;
    const int lane = tid & 31;
    const int nl   = lane & 15;
    const int hi   = lane >> 4;    // 0 or 1
    const int kk   = 2 * hi;       // K sub-offset inside each 4-wide K step

    for (int ntile = wave; ntile < NTILES; ntile += 8) {
        const int n = ntile * 16 + nl;

        // Preload this lane's B fragments for all 16 K-steps (32 VGPRs).
        const float* wrow = proj_w + (size_t)n * CDIM + kk;
        v2f bf[16];
        #pragma unroll
        for (int st = 0; st < 16; ++st)
            bf[st] = *(const v2f*)(wrow + 4 * st);

        const float pbias = pb[n];

        for (int mtile = 0; mtile < MTILES; ++mtile) {
            const int    m = mtile * 16 + nl;   // this lane's A row
            const float4 x = xs[m];             // (q, attempts, mastery, valid)

            v8f acc = {0.f, 0.f, 0.f, 0.f, 0.f, 0.f, 0.f, 0.f};

            #pragma unroll
            for (int st = 0; st < 16; ++st) {
                const int k0 = 4 * st + kk;
                // Segment select: k<21 -> skill, k<42 -> attempts, else mastery.
                const float xa = (k0     < 21) ? x.x : ((k0     < 42) ? x.y : x.z);
                const float xb = (k0 + 1 < 21) ? x.x : ((k0 + 1 < 42) ? x.y : x.z);
                v2f a;
                a.x = xa * wc[k0]     + bc[k0];
                a.y = xb * wc[k0 + 1] + bc[k0 + 1];
                acc = __builtin_amdgcn_wmma_f32_16x16x4_f32(
                          false, a, false, bf[st], (short)0, acc, false, false);
            }

            // Store the 16x16 tile: add bias, scale by valid flag, mask s >= SEQ.
            #pragma unroll
            for (int v = 0; v < 8; ++v) {
                const int s = mtile * 16 + v + 8 * hi;
                if (s < SEQ) {
                    const float val = (acc[v] + pbias) * xs[s].w;
                    out[((size_t)b * SEQ + s) * EMB + n] = val;
                }
            }
        }
    }
}

extern "C" void kernel_launch(void* const* d_in, const int* in_sizes, int n_in,
                              void* d_out, int out_size, void* d_ws, size_t ws_size,
                              hipStream_t stream)
{
    // setup_inputs() order:
    // 0:q 1:r 2:qry(unused) 3:skill_w 4:skill_b 5:att_w 6:att_b
    // 7:mast_w 8:mast_b 9:proj_w 10:proj_b
    const int*   q       = (const int*)  d_in[0];
    const int*   r       = (const int*)  d_in[1];
    const float* skill_w = (const float*)d_in[3];
    const float* skill_b = (const float*)d_in[4];
    const float* att_w   = (const float*)d_in[5];
    const float* att_b   = (const float*)d_in[6];
    const float* mast_w  = (const float*)d_in[7];
    const float* mast_b  = (const float*)d_in[8];
    const float* proj_w  = (const float*)d_in[9];
    const float* proj_b  = (const float*)d_in[10];
    float*       out     = (float*)d_out;

    dim3 grid(BATCH);
    dim3 block(256);
    traj_encoder_wmma_kernel<<<grid, block, 0, stream>>>(
        q, r, skill_w, skill_b, att_w, att_b, mast_w, mast_b, proj_w, proj_b, out);
}